// FieldAwareFactorizationMachineLayer_25872882991599
// MI455X (gfx1250) — compile-verified
//
#include <hip/hip_runtime.h>
#include <cstdint>
#include <cstddef>

// Problem constants (match reference)
#define NF     20u                // NUM_FIELDS
#define EMB    64u                // EMBED
#define NBATCH 4096u              // BATCH
#define NPAIR  190u               // NF*(NF-1)/2
#define ROWS   (NBATCH * NPAIR)   // 778240 output rows of 64 floats
#define RPB    16u                // rows per block-iteration (256 thr / 16 lanes-per-row)
#define NGRP   (ROWS / RPB)       // 48640 row-groups
#define GRID   4864               // NGRP / GRID == 10 iterations per block
#define EV4    (EMB / 4u)         // float4 slots per row

typedef __attribute__((ext_vector_type(4))) float floatx4;
// Async builtin takes pointers to a 16-byte int vector: AS(1) global source,
// AS(3) LDS destination.
typedef int v4i __attribute__((vector_size(4 * sizeof(int))));
typedef __attribute__((address_space(1))) v4i* gas_v4;   // global
typedef __attribute__((address_space(3))) v4i* lds_v4;   // LDS

#if __has_builtin(__builtin_amdgcn_global_load_async_to_lds_b128)
#define HAVE_ASYNC 1
#else
#define HAVE_ASYNC 0
#endif

#if __has_builtin(__builtin_amdgcn_s_wait_asynccnt)
#define WAIT_ASYNC(n) __builtin_amdgcn_s_wait_asynccnt(n)
#else
#define WAIT_ASYNC(n) asm volatile("s_wait_asynccnt %0" :: "i"(n) : "memory")
#endif

// Raw hardware sqrt (v_sqrt_f32, <=1 ulp) instead of the correctly-rounded
// software expansion.
#if __has_builtin(__builtin_amdgcn_sqrtf)
#define RAW_SQRTF(x) __builtin_amdgcn_sqrtf(x)
#else
#define RAW_SQRTF(x) __builtin_sqrtf(x)
#endif

// Closed-form upper-triangle pair index p -> (i, j), i < j, row-major order.
// off(i) = (39i - i^2)/2. f = (39-s)/2 - i is 0 at segment boundaries and in
// [0.051, 0.946] elsewhere; a 1-ulp sqrt error moves f by ~1e-6, so adding
// eps = 1/64 before flooring is exact for all p in [0, 190).
__device__ __forceinline__ void pair_ij(uint32_t p, uint32_t& pi, uint32_t& pj) {
    const float s = RAW_SQRTF((float)(1521u - 8u * p));
    const uint32_t i = (uint32_t)(int)((39.0f - s) * 0.5f + 0.015625f);
    const uint32_t rem = p - ((39u * i - i * i) >> 1);
    pi = i;
    pj = i + 1u + rem;
}

// Element offsets (uint32, whole tensor < 2^27 elements) for output row `row`,
// float4 slot `ev`: offA -> v[b,j,i,e], offB -> v[b,i,j,e].
__device__ __forceinline__ void row_offsets(uint32_t row, uint32_t ev,
                                            uint32_t& offA, uint32_t& offB) {
    const uint32_t b = row / NPAIR;          // unsigned magic-mul division
    const uint32_t p = row - b * NPAIR;
    uint32_t i, j;
    pair_ij(p, i, j);
    const uint32_t e = ev << 2;
    offA = ((b * NF + j) * NF + i) * EMB + e;   // v[b, j, i, e]
    offB = ((b * NF + i) * NF + j) * EMB + e;   // v[b, i, j, e]
}

__global__ __launch_bounds__(256)
void ffm_pair_mul(const float* __restrict__ in, float* __restrict__ out) {
    const uint32_t tid = threadIdx.x;
    const uint32_t rl  = tid >> 4;     // row within group (0..15)
    const uint32_t ev  = tid & 15u;    // float4 slot within row (0..15)

#if HAVE_ASYNC
    // Double-buffered async staging: 2 buffers * 16 rows * 2 operands * 256B = 16 KB LDS.
    __shared__ floatx4 lA[2][RPB][EV4];
    __shared__ floatx4 lB[2][RPB][EV4];

    uint32_t g = blockIdx.x;

    // Prologue: stage group g into buffer 0 (2 async b128 per wave).
    {
        uint32_t offA, offB;
        row_offsets(g * RPB + rl, ev, offA, offB);
        __builtin_amdgcn_global_load_async_to_lds_b128(
            (gas_v4)(in + offA), (lds_v4)&lA[0][rl][ev], 0, 0);
        __builtin_amdgcn_global_load_async_to_lds_b128(
            (gas_v4)(in + offB), (lds_v4)&lB[0][rl][ev], 0, 0);
    }

    uint32_t buf = 0;
    while (g < NGRP) {
        const uint32_t gn = g + gridDim.x;
        if (gn < NGRP) {
            // Stage next group into the other buffer, then retire only the
            // previous stage (async loads complete in order): wait cnt<=2.
            uint32_t offA, offB;
            row_offsets(gn * RPB + rl, ev, offA, offB);
            __builtin_amdgcn_global_load_async_to_lds_b128(
                (gas_v4)(in + offA), (lds_v4)&lA[buf ^ 1u][rl][ev], 0, 0);
            __builtin_amdgcn_global_load_async_to_lds_b128(
                (gas_v4)(in + offB), (lds_v4)&lB[buf ^ 1u][rl][ev], 0, 0);
            WAIT_ASYNC(2);
        } else {
            WAIT_ASYNC(0);
        }

        // Multiply current group out of LDS (each lane reads what it wrote).
        {
            const uint32_t oofs = (g * RPB + rl) * EMB + (ev << 2);
            const floatx4 a = lA[buf][rl][ev];
            const floatx4 b = lB[buf][rl][ev];
            const floatx4 r = a * b;
            // Write-once output: non-temporal, don't pollute the 192MB L2.
            __builtin_nontemporal_store(r, (floatx4*)(out + oofs));
        }

        buf ^= 1u;
        g = gn;
    }
#else
    // Fallback: direct coalesced b128 gather-multiply-store.
    for (uint32_t g = blockIdx.x; g < NGRP; g += gridDim.x) {
        uint32_t offA, offB;
        row_offsets(g * RPB + rl, ev, offA, offB);
        const uint32_t oofs = (g * RPB + rl) * EMB + (ev << 2);
        const floatx4 a = *(const floatx4*)(in + offA);
        const floatx4 b = *(const floatx4*)(in + offB);
        __builtin_nontemporal_store(a * b, (floatx4*)(out + oofs));
    }
#endif
}

extern "C" void kernel_launch(void* const* d_in, const int* in_sizes, int n_in,
                              void* d_out, int out_size, void* d_ws, size_t ws_size,
                              hipStream_t stream) {
    (void)in_sizes; (void)n_in; (void)out_size; (void)d_ws; (void)ws_size;
    const float* in = (const float*)d_in[0];
    float* out = (float*)d_out;
    ffm_pair_mul<<<dim3(GRID), dim3(256), 0, stream>>>(in, out);
}